// MHAGRU_37658273251688
// MI455X (gfx1250) — compile-verified
//
#include <hip/hip_runtime.h>
#include <hip/hip_bf16.h>

// Problem constants (match reference)
#define BB 32
#define TT 64
#define FF 128
#define HH 8
#define HID 32

typedef __attribute__((ext_vector_type(2))) float v2f;
typedef __attribute__((ext_vector_type(8))) float v8f;

// ---------------------------------------------------------------------------
// FP32 WMMA GEMM:  C[M,N] = A[M,K] @ Wt[N,K]^T + bias[N]
// One wave (32 threads) computes one 16x16 tile using V_WMMA_F32_16X16X4_F32.
// A-matrix layout: lane l holds A[m0 + (l&15)][k + 2*(l>>4) + j], j=0,1
// B-matrix layout: lane l holds B[k + 2*(l>>4) + j][n0 + (l&15)]  (B[k][n]=Wt[n][k])
// C/D layout:      lane l, vgpr j -> C[m0 + j + 8*(l>>4)][n0 + (l&15)]
// ---------------------------------------------------------------------------
__global__ void gemm_wmma_f32(const float* __restrict__ A,
                              const float* __restrict__ Wt,
                              const float* __restrict__ bias,
                              float* __restrict__ C,
                              int M, int N, int K) {
    const int l    = threadIdx.x;          // 0..31
    const int lm   = l & 15;
    const int half = l >> 4;               // 0 or 1
    const int m0   = blockIdx.y * 16;
    const int n0   = blockIdx.x * 16;

    const float bval = bias[n0 + lm];
    v8f c;
#pragma unroll
    for (int j = 0; j < 8; ++j) c[j] = bval;

    const float* arow = A + (size_t)(m0 + lm) * K;
    const float* brow = Wt + (size_t)(n0 + lm) * K;

    for (int k = 0; k < K; k += 4) {
        const int ko = k + 2 * half;
        v2f a, b;
        a[0] = arow[ko];
        a[1] = arow[ko + 1];
        b[0] = brow[ko];
        b[1] = brow[ko + 1];
        c = __builtin_amdgcn_wmma_f32_16x16x4_f32(
                /*neg_a=*/false, a, /*neg_b=*/false, b,
                /*c_mod=*/(short)0, c, /*reuse_a=*/false, /*reuse_b=*/false);
    }

#pragma unroll
    for (int j = 0; j < 8; ++j)
        C[(size_t)(m0 + j + 8 * half) * N + (n0 + lm)] = c[j];
}

// ---------------------------------------------------------------------------
// Per-feature GRU scan over T. One wave per feature f; lane = batch b.
// Weights for f are wave-uniform: W_hh staged in LDS, small vectors in regs.
// xp layout [B*T, F]; out layout [B, T, F, H] (== [B*T, F*H] rows for out_proj)
// ---------------------------------------------------------------------------
__global__ void gru_scan(const float* __restrict__ xp,
                         const float* __restrict__ W_ih,
                         const float* __restrict__ b_ih,
                         const float* __restrict__ W_hh,
                         const float* __restrict__ b_hh,
                         float* __restrict__ out) {
    const int f = blockIdx.x;     // 0..F-1
    const int b = threadIdx.x;    // 0..31 (== B lanes)

    __shared__ float sWhh[3 * HH * HH];  // 192 floats: W_hh[f, g, h]
    for (int i = threadIdx.x; i < 3 * HH * HH; i += 32)
        sWhh[i] = W_hh[(size_t)f * 3 * HH * HH + i];
    __syncthreads();

    float wih[3 * HH], bih[3 * HH], bhh[3 * HH];
#pragma unroll
    for (int j = 0; j < 3 * HH; ++j) {
        wih[j] = W_ih[(size_t)f * 3 * HH + j];
        bih[j] = b_ih[(size_t)f * 3 * HH + j];
        bhh[j] = b_hh[(size_t)f * 3 * HH + j];
    }

    float h[HH];
#pragma unroll
    for (int i = 0; i < HH; ++i) h[i] = 0.0f;

    for (int t = 0; t < TT; ++t) {
        const float xv = xp[(size_t)(b * TT + t) * FF + f];

        float gh[3 * HH];
#pragma unroll
        for (int g = 0; g < 3 * HH; ++g) {
            float acc = bhh[g];
#pragma unroll
            for (int k = 0; k < HH; ++k)
                acc = fmaf(h[k], sWhh[g * HH + k], acc);
            gh[g] = acc;
        }

        float hn[HH];
#pragma unroll
        for (int i = 0; i < HH; ++i) {
            const float gx_r = fmaf(xv, wih[i],          bih[i]);
            const float gx_z = fmaf(xv, wih[HH + i],     bih[HH + i]);
            const float gx_n = fmaf(xv, wih[2 * HH + i], bih[2 * HH + i]);
            const float r = 1.0f / (1.0f + __expf(-(gx_r + gh[i])));
            const float z = 1.0f / (1.0f + __expf(-(gx_z + gh[HH + i])));
            const float na = gx_n + r * gh[2 * HH + i];
            const float e2 = __expf(2.0f * na);                // tanh via exp
            const float n  = (e2 - 1.0f) / (e2 + 1.0f);
            hn[i] = (1.0f - z) * n + z * h[i];
        }

        float* dst = out + ((size_t)(b * TT + t) * FF + f) * HH;
        float4* d4 = (float4*)dst;
        d4[0] = make_float4(hn[0], hn[1], hn[2], hn[3]);
        d4[1] = make_float4(hn[4], hn[5], hn[6], hn[7]);
#pragma unroll
        for (int i = 0; i < HH; ++i) h[i] = hn[i];
    }
}

// ---------------------------------------------------------------------------
// Shared MHA over time for each (b,f) sequence. One 64-thread block per
// sequence; thread t owns query row t. K/V staged in LDS (broadcast reads).
// ---------------------------------------------------------------------------
__global__ void mha_seq(const float* __restrict__ gru,   // [B,T,F,H]
                        const float* __restrict__ in_w,  // [3H,H]
                        const float* __restrict__ in_b,  // [3H]
                        const float* __restrict__ ow,    // [H,H]
                        const float* __restrict__ ob,    // [H]
                        float* __restrict__ att) {       // [B,T,F,H]
    const int f = blockIdx.x;
    const int b = blockIdx.y;
    const int t = threadIdx.x;   // 0..63

    __shared__ float kk[TT][HH];
    __shared__ float vv[TT][HH];

    float s[HH];
    const float* src = gru + ((size_t)(b * TT + t) * FF + f) * HH;
#pragma unroll
    for (int i = 0; i < HH; ++i) s[i] = src[i];

    float q[HH];
#pragma unroll
    for (int j = 0; j < HH; ++j) {
        float aq = in_b[j], ak = in_b[HH + j], av = in_b[2 * HH + j];
#pragma unroll
        for (int k = 0; k < HH; ++k) {
            aq = fmaf(s[k], in_w[j * HH + k], aq);
            ak = fmaf(s[k], in_w[(HH + j) * HH + k], ak);
            av = fmaf(s[k], in_w[(2 * HH + j) * HH + k], av);
        }
        q[j] = aq;
        kk[t][j] = ak;
        vv[t][j] = av;
    }
    __syncthreads();

    const float scale = 0.70710678118654752f;   // 1/sqrt(head_dim=2)
    float ctx[HH];
#pragma unroll
    for (int n = 0; n < 4; ++n) {               // NH=4 heads, HD=2
        const float q0 = q[2 * n] * scale;
        const float q1 = q[2 * n + 1] * scale;
        float mx = -1e30f;
        for (int j = 0; j < TT; ++j) {
            const float sc = fmaf(q0, kk[j][2 * n], q1 * kk[j][2 * n + 1]);
            mx = fmaxf(mx, sc);
        }
        float ssum = 0.0f, c0 = 0.0f, c1 = 0.0f;
        for (int j = 0; j < TT; ++j) {
            const float sc = fmaf(q0, kk[j][2 * n], q1 * kk[j][2 * n + 1]);
            const float e = __expf(sc - mx);
            ssum += e;
            c0 = fmaf(e, vv[j][2 * n], c0);
            c1 = fmaf(e, vv[j][2 * n + 1], c1);
        }
        const float inv = 1.0f / ssum;
        ctx[2 * n] = c0 * inv;
        ctx[2 * n + 1] = c1 * inv;
    }

    float o[HH];
#pragma unroll
    for (int j = 0; j < HH; ++j) {
        float a = ob[j];
#pragma unroll
        for (int k = 0; k < HH; ++k) a = fmaf(ctx[k], ow[j * HH + k], a);
        o[j] = a;
    }

    float* dst = att + ((size_t)(b * TT + t) * FF + f) * HH;
    float4* d4 = (float4*)dst;
    d4[0] = make_float4(o[0], o[1], o[2], o[3]);
    d4[1] = make_float4(o[4], o[5], o[6], o[7]);
}

// ---------------------------------------------------------------------------
extern "C" void kernel_launch(void* const* d_in, const int* in_sizes, int n_in,
                              void* d_out, int out_size, void* d_ws, size_t ws_size,
                              hipStream_t stream) {
    const float* x    = (const float*)d_in[0];   // [B,T,F]
    const float* Wp   = (const float*)d_in[1];   // [F,F]
    const float* bp   = (const float*)d_in[2];   // [F]
    const float* W_ih = (const float*)d_in[3];   // [F,3H]
    const float* b_ih = (const float*)d_in[4];   // [F,3H]
    const float* W_hh = (const float*)d_in[5];   // [F,3H,H]
    const float* b_hh = (const float*)d_in[6];   // [F,3H]
    const float* in_w = (const float*)d_in[7];   // [3H,H]
    const float* in_b = (const float*)d_in[8];   // [3H]
    const float* ow   = (const float*)d_in[9];   // [H,H]
    const float* ob   = (const float*)d_in[10];  // [H]
    const float* Wout = (const float*)d_in[11];  // [HID, F*H]
    const float* bout = (const float*)d_in[12];  // [HID]

    float* xp  = (float*)d_ws;                         // [B*T, F]  = 2048x128
    float* gru = xp + (size_t)BB * TT * FF;            // [B,T,F,H] = 2048x1024

    float* y   = (float*)d_out;                        // [B,T,HID] = 65536
    float* att = y + (size_t)BB * TT * HID;            // [B,T,F,H]

    // 1) input_proj: xp = x @ Wp^T + bp  (M=2048, N=128, K=128), WMMA f32
    {
        dim3 grid(FF / 16, (BB * TT) / 16);
        gemm_wmma_f32<<<grid, 32, 0, stream>>>(x, Wp, bp, xp,
                                               BB * TT, FF, FF);
    }

    // 2) per-feature GRU scan (one wave per feature, lane = batch)
    gru_scan<<<FF, 32, 0, stream>>>(xp, W_ih, b_ih, W_hh, b_hh, gru);

    // 3) shared MHA per (b,f) sequence
    {
        dim3 grid(FF, BB);
        mha_seq<<<grid, TT, 0, stream>>>(gru, in_w, in_b, ow, ob, att);
    }

    // 4) out_proj: y = gru[2048,1024] @ Wout^T + bout (M=2048, N=32, K=1024)
    {
        dim3 grid(HID / 16, (BB * TT) / 16);
        gemm_wmma_f32<<<grid, 32, 0, stream>>>(gru, Wout, bout, y,
                                               BB * TT, HID, FF * HH);
    }
}